// Model_35029753266584
// MI455X (gfx1250) — compile-verified
//
#include <hip/hip_runtime.h>
#include <hip/hip_bf16.h>

#define USERS 50000
#define ITEMS 50000
#define NN    (USERS + ITEMS)
#define DD    64
#define NE    3200000
#define LEAK  0.2f
#define RIS_ADJ 0.2f
#define RIS   0.5f

typedef __attribute__((ext_vector_type(2))) float v2f;
typedef __attribute__((ext_vector_type(8))) float v8f;

// ---- ISA-native f32 atomic add, no-return form (STOREcnt), device scope ----
__device__ __forceinline__ void atomAddF32(float* p, float v) {
  asm volatile("global_atomic_add_f32 %0, %1, off scope:SCOPE_DEV"
               :: "v"(p), "v"(v)
               : "memory");
}

// ---- float <-> order-preserving uint encoding for atomicMax on floats ----
__device__ __forceinline__ unsigned fenc(float f) {
  unsigned u = __float_as_uint(f);
  return (u & 0x80000000u) ? ~u : (u | 0x80000000u);
}
__device__ __forceinline__ float fdec(unsigned u) {
  return (u & 0x80000000u) ? __uint_as_float(u & 0x7fffffffu) : __uint_as_float(~u);
}

// ============================================================================
// GEMM: out[M,64] = X[M,K] @ W[K,64] + b, via V_WMMA_F32_16X16X4_F32.
// One wave -> one 16x16 output tile; block (32,4) -> 16x64 row panel.
// A tile 16x4 f32: lanes 0-15 hold M=0..15 (K=k0,k0+1), lanes 16-31 K=k0+2,k0+3.
// B tile 4x16 f32: lane n holds N=n; vgpr0/1 = K rows (k0,k0+1 | k0+2,k0+3).
// D tile: vgpr r -> row m0 + r + 8*half, col n0 + (lane&15).
// ============================================================================
__global__ void __launch_bounds__(128)
mmgnn_gemm_bias(const float* __restrict__ X, const float* __restrict__ W,
                const float* __restrict__ b, float* __restrict__ out, int K) {
  const int lane = threadIdx.x;
  const int l16  = lane & 15;
  const int half = lane >> 4;
  const int n0   = threadIdx.y * 16;
  const int m0   = blockIdx.x * 16;

  const float* xr = X + (size_t)(m0 + l16) * K + 2 * half;   // row-major A
  const float* wc = W + (size_t)(2 * half) * DD + n0 + l16;  // row-major B
  v8f acc = {};
#pragma unroll 4
  for (int k = 0; k < K; k += 4) {
    v2f a, bb;
    a.x  = xr[k];
    a.y  = xr[k + 1];
    bb.x = wc[(size_t)k * DD];
    bb.y = wc[(size_t)(k + 1) * DD];
    acc = __builtin_amdgcn_wmma_f32_16x16x4_f32(
        /*neg_a=*/false, a, /*neg_b=*/false, bb,
        /*c_mod=*/(short)0, acc, /*reuse_a=*/false, /*reuse_b=*/false);
  }
  const float bias = b[n0 + l16];
  float* dst = out + (size_t)(m0 + 8 * half) * DD + n0 + l16;
#pragma unroll
  for (int r = 0; r < 8; ++r) dst[(size_t)r * DD] = acc[r] + bias;
}

// ---- in-place row l2-normalize, one wave per row (float2 per lane) ----
__global__ void __launch_bounds__(256)
mmgnn_l2norm_rows(float* __restrict__ x, int nrows) {
  const int row = blockIdx.x * blockDim.y + threadIdx.y;
  if (row >= nrows) return;
  const int lane = threadIdx.x;
  float2* p = (float2*)(x + (size_t)row * DD);
  float2 v = p[lane];
  float s = v.x * v.x + v.y * v.y;
  for (int o = 16; o > 0; o >>= 1) s += __shfl_xor(s, o, 32);
  const float inv = 1.0f / fmaxf(sqrtf(s), 1e-12f);
  v.x *= inv; v.y *= inv;
  p[lane] = v;
}

// ============================================================================
// SpMM: out[rows[e]] += vals[e] * x[cols[e]].  Dense operand is a virtual
// concat: col<USERS -> xu[col], else xi[col-USERS].  One wave per edge:
// 32 lanes x float2 = 64 dims; coalesced 256B gather (L2-resident) +
// global_atomic_add_f32 scatter.  Edge streams use NT loads so the 192MB L2
// keeps the dense gather/scatter working set instead of stream-once triples.
// ============================================================================
__global__ void __launch_bounds__(256)
mmgnn_spmm(const int* __restrict__ rows, const int* __restrict__ cols,
           const float* __restrict__ vals, int nnz,
           const float* __restrict__ xu, const float* __restrict__ xi,
           float* __restrict__ out) {
  const int lane   = threadIdx.x & 31;
  const int wid    = (blockIdx.x * blockDim.x + threadIdx.x) >> 5;
  const int nwaves = (gridDim.x * blockDim.x) >> 5;
  for (int base = wid * 32; base < nnz; base += nwaves * 32) {
    const int e = base + lane;
    int r = 0, c = 0;
    float v = 0.0f;
    if (e < nnz) {
      r = __builtin_nontemporal_load(rows + e);
      c = __builtin_nontemporal_load(cols + e);
      v = __builtin_nontemporal_load(vals + e);
    }
    // prefetch next edge chunk for this wave (global_prefetch_b8)
    const int nxt = base + nwaves * 32 + lane;
    if (nxt < nnz) {
      __builtin_prefetch(rows + nxt, 0, 0);
      __builtin_prefetch(cols + nxt, 0, 0);
      __builtin_prefetch(vals + nxt, 0, 0);
    }
    const int cnt = (nnz - base < 32) ? (nnz - base) : 32;
    for (int j = 0; j < cnt; ++j) {
      const int   rj = __shfl(r, j, 32);
      const int   cj = __shfl(c, j, 32);
      const float vj = __shfl(v, j, 32);
      const float2* src = (const float2*)((cj < USERS)
                              ? (xu + (size_t)cj * DD)
                              : (xi + (size_t)(cj - USERS) * DD));
      const float2 xv = src[lane];
      float* dst = out + (size_t)rj * DD + lane * 2;
      atomAddF32(dst,     xv.x * vj);
      atomAddF32(dst + 1, xv.y * vj);
    }
  }
}

// ---- fuse modal mixing: modal = w0*(base_img+Sadj+0.2*Simg) + w1*(T1+T2+0.2*Stxt)
//      also seeds embeds and total(=d_out). ----
__global__ void __launch_bounds__(256)
mmgnn_combine(const float* __restrict__ u, const float* __restrict__ nimg,
              const float* __restrict__ Sadj, const float* __restrict__ Simg,
              const float* __restrict__ T1, const float* __restrict__ T2,
              const float* __restrict__ Stxt, const float* __restrict__ mw,
              float* __restrict__ modal, float* __restrict__ embeds,
              float* __restrict__ total) {
  const size_t idx = (size_t)blockIdx.x * blockDim.x + threadIdx.x;
  if (idx >= (size_t)NN * DD) return;
  const float m0 = mw[0], m1 = mw[1];
  const float mx = fmaxf(m0, m1);
  const float e0 = __expf(m0 - mx), e1 = __expf(m1 - mx);
  const float w0 = e0 / (e0 + e1), w1 = e1 / (e0 + e1);
  const float base = (idx < (size_t)USERS * DD) ? u[idx] : nimg[idx - (size_t)USERS * DD];
  const float img = base + Sadj[idx] + RIS_ADJ * Simg[idx];
  const float txt = T1[idx] + T2[idx] + RIS_ADJ * Stxt[idx];
  const float m = w0 * img + w1 * txt;
  modal[idx] = m; embeds[idx] = m; total[idx] = m;
}

__global__ void mmgnn_init_reduce(unsigned* red) {
  red[0] = fenc(-__builtin_inff());
  red[1] = __float_as_uint(0.0f);
}

// ---- scores[i] = E[i,:] . w ; track global max (encoded atomicMax) ----
__global__ void __launch_bounds__(256)
mmgnn_scores_max(const float* __restrict__ E, const float* __restrict__ w,
                 float* __restrict__ scores, unsigned* __restrict__ red) {
  __shared__ float smax[8];
  const int row  = blockIdx.x * blockDim.y + threadIdx.y;
  const int lane = threadIdx.x;
  float s = -__builtin_inff();
  if (row < NN) {
    const float2 ev = ((const float2*)(E + (size_t)row * DD))[lane];
    const float2 wv = ((const float2*)w)[lane];
    float p = ev.x * wv.x + ev.y * wv.y;
    for (int o = 16; o > 0; o >>= 1) p += __shfl_xor(p, o, 32);
    if (lane == 0) scores[row] = p;
    s = p;
  }
  if (lane == 0) smax[threadIdx.y] = s;
  __syncthreads();
  if (threadIdx.x == 0 && threadIdx.y == 0) {
    float m = smax[0];
    for (int i = 1; i < 8; ++i) m = fmaxf(m, smax[i]);
    atomicMax(red, fenc(m));
  }
}

__global__ void __launch_bounds__(256)
mmgnn_sum_exp(const float* __restrict__ scores, unsigned* __restrict__ red) {
  __shared__ float sh[256];
  const float mx = fdec(red[0]);
  const int i = blockIdx.x * 256 + threadIdx.x;
  sh[threadIdx.x] = (i < NN) ? __expf(scores[i] - mx) : 0.0f;
  __syncthreads();
  for (int o = 128; o > 0; o >>= 1) {
    if (threadIdx.x < o) sh[threadIdx.x] += sh[threadIdx.x + o];
    __syncthreads();
  }
  if (threadIdx.x == 0) atomAddF32((float*)(red + 1), sh[0]);
}

// ---- embeds = leaky_relu(E * att) ; total += embeds ----
__global__ void __launch_bounds__(256)
mmgnn_apply_att(const float* __restrict__ E, const float* __restrict__ scores,
                const unsigned* __restrict__ red, float* __restrict__ embeds,
                float* __restrict__ total) {
  const size_t idx = (size_t)blockIdx.x * blockDim.x + threadIdx.x;
  if (idx >= (size_t)NN * DD) return;
  const float mx  = fdec(red[0]);
  const float sum = __uint_as_float(red[1]);
  const int row = (int)(idx >> 6);  // DD == 64
  const float att = __expf(scores[row] - mx) / sum;
  float v = E[idx] * att;
  v = (v >= 0.0f) ? v : LEAK * v;
  embeds[idx] = v;
  total[idx] += v;
}

// ---- out += RIS * l2norm_rows(modal) ----
__global__ void __launch_bounds__(256)
mmgnn_final_add(const float* __restrict__ modal, float* __restrict__ out) {
  const int row = blockIdx.x * blockDim.y + threadIdx.y;
  if (row >= NN) return;
  const int lane = threadIdx.x;
  const float2 m = ((const float2*)(modal + (size_t)row * DD))[lane];
  float s = m.x * m.x + m.y * m.y;
  for (int o = 16; o > 0; o >>= 1) s += __shfl_xor(s, o, 32);
  const float sc = RIS / fmaxf(sqrtf(s), 1e-12f);
  float2* o2 = (float2*)(out + (size_t)row * DD) + lane;
  float2 cur = *o2;
  cur.x += m.x * sc; cur.y += m.y * sc;
  *o2 = cur;
}

extern "C" void kernel_launch(void* const* d_in, const int* in_sizes, int n_in,
                              void* d_out, int out_size, void* d_ws, size_t ws_size,
                              hipStream_t stream) {
  const int*   adj_r = (const int*)d_in[0];
  const int*   adj_c = (const int*)d_in[1];
  const float* adj_v = (const float*)d_in[2];
  const int*   img_r = (const int*)d_in[3];
  const int*   img_c = (const int*)d_in[4];
  const float* img_v = (const float*)d_in[5];
  const int*   txt_r = (const int*)d_in[6];
  const int*   txt_c = (const int*)d_in[7];
  const float* txt_v = (const float*)d_in[8];
  const float* u     = (const float*)d_in[9];
  const float* it    = (const float*)d_in[10];
  const float* img_W = (const float*)d_in[11];
  const float* img_b = (const float*)d_in[12];
  const float* txt_W = (const float*)d_in[13];
  const float* txt_b = (const float*)d_in[14];
  const float* mw    = (const float*)d_in[15];
  const float* attw  = (const float*)d_in[16];
  const float* imgE  = (const float*)d_in[17];  // [USERS,1024]
  const float* txtE  = (const float*)d_in[18];  // [USERS,384]
  float* out = (float*)d_out;

  float* ws = (float*)d_ws;
  const size_t NB = (size_t)NN * DD;
  float* Sadj   = ws;                 // also reused as E per GNN layer
  float* Simg   = ws + 1 * NB;
  float* Stxt   = ws + 2 * NB;
  float* T1     = ws + 3 * NB;
  float* T2     = ws + 4 * NB;
  float* modal  = ws + 5 * NB;
  float* embeds = ws + 6 * NB;
  float* fimg   = ws + 7 * NB;                      // [USERS,64]
  float* ftxt   = fimg + (size_t)USERS * DD;        // [USERS,64]
  float* scores = ftxt + (size_t)USERS * DD;        // [NN]
  unsigned* red = (unsigned*)(scores + NN);         // [2]: enc(max), sum

  // zero all atomic-accumulated SpMM outputs
  hipMemsetAsync(Sadj, 0, 5 * NB * sizeof(float), stream);

  // dense projections (WMMA f32) + row l2norm in place
  dim3 gblk(32, 4);
  mmgnn_gemm_bias<<<USERS / 16, gblk, 0, stream>>>(imgE, img_W, img_b, fimg, 1024);
  mmgnn_gemm_bias<<<USERS / 16, gblk, 0, stream>>>(txtE, txt_W, txt_b, ftxt, 384);
  dim3 rblk(32, 8);
  mmgnn_l2norm_rows<<<(USERS + 7) / 8, rblk, 0, stream>>>(fimg, USERS);
  mmgnn_l2norm_rows<<<(USERS + 7) / 8, rblk, 0, stream>>>(ftxt, USERS);

  const int SPB = 1024;
  mmgnn_spmm<<<SPB, 256, 0, stream>>>(adj_r, adj_c, adj_v, NE, u, it, Sadj);   // spmm(adj,e0)
  mmgnn_spmm<<<SPB, 256, 0, stream>>>(img_r, img_c, img_v, NE, u, it, Simg);   // spmm(img,e0)
  mmgnn_spmm<<<SPB, 256, 0, stream>>>(txt_r, txt_c, txt_v, NE, u, it, Stxt);   // spmm(txt,e0)
  mmgnn_spmm<<<SPB, 256, 0, stream>>>(adj_r, adj_c, adj_v, NE, u, ftxt, T1);   // T1
  mmgnn_spmm<<<SPB, 256, 0, stream>>>(adj_r, adj_c, adj_v, NE, T1, it, T2);    // T2

  const int eblocks = (int)((NB + 255) / 256);
  mmgnn_combine<<<eblocks, 256, 0, stream>>>(u, fimg, Sadj, Simg, T1, T2, Stxt,
                                             mw, modal, embeds, out);

  for (int l = 0; l < 2; ++l) {
    hipMemsetAsync(Sadj, 0, NB * sizeof(float), stream);
    mmgnn_spmm<<<SPB, 256, 0, stream>>>(adj_r, adj_c, adj_v, NE,
                                        embeds, embeds + (size_t)USERS * DD, Sadj);
    mmgnn_init_reduce<<<1, 1, 0, stream>>>(red);
    mmgnn_scores_max<<<(NN + 7) / 8, rblk, 0, stream>>>(Sadj, attw + (size_t)l * DD, scores, red);
    mmgnn_sum_exp<<<(NN + 255) / 256, 256, 0, stream>>>(scores, red);
    mmgnn_apply_att<<<eblocks, 256, 0, stream>>>(Sadj, scores, red, embeds, out);
  }

  mmgnn_final_add<<<(NN + 7) / 8, rblk, 0, stream>>>(modal, out);
}